// ExceptionalEGNN_85048942395859
// MI455X (gfx1250) — compile-verified
//
#include <hip/hip_runtime.h>
#include <math.h>

// ---------------------------------------------------------------------------
// ExceptionalEGNN for MI455X (gfx1250, wave32).
// Dims: N=10000 nodes, E=160000 edges, IN=128, H=256, D=248 (pad->256),
// L=2 layers, NNZ=512 structure constants, G=16 graphs, OUT=16.
// All GEMMs run on V_WMMA_F32_16X16X4_F32 (exact fp32, matrix pipe).
// Brackets: 1 edge per wave, LDS-staged h rows + structure constants,
// ds_add_f32 local accumulation, global_atomic_add_f32 scatter.
// ---------------------------------------------------------------------------

#define N_NODES 10000
#define N_EDGES 160000
#define IN_DIM  128
#define H_DIM   256
#define A_DIM   248      // algebra dim D
#define DP      256      // padded D
#define LDH     512      // leading dim of [h | agg] buffer
#define N_LAYERS 2
#define NNZ_C   512
#define N_GRAPHS 16
#define OUT_DIM  16

typedef __attribute__((ext_vector_type(2))) float v2f;
typedef __attribute__((ext_vector_type(8))) float v8f;

// ---------------------------------------------------------------------------
// WMMA f32 GEMM:  C[M,Ntiles*16] = act(A[M,K] @ Bfrag + bias) (+ res)
// 1 wave -> one 16x16 tile.  block = (32,4) waves, grid = (M/16, ceil(Nt/4)).
// A-fragment (16x4 f32): lane row = lane&15, K pair = (lane>>4)*2 + {0,1}.
// B-fragment pre-swizzled: contiguous v2f per (k4, ntile, lane).
// C layout: vgpr r, lane l -> row = (l>>4)*8 + r, col = l&15.
// ---------------------------------------------------------------------------
__global__ __launch_bounds__(128)
void wmma_gemm_f32(const float* __restrict__ A, int lda,
                   const float* __restrict__ Bf,
                   const float* __restrict__ bias,
                   const float* __restrict__ res, int ldr,
                   float* __restrict__ C, int ldc,
                   int K4, int Ntiles, int act)
{
    const int ntile = blockIdx.y * blockDim.y + threadIdx.y;
    if (ntile >= Ntiles) return;                  // whole-wave exit, EXEC stays full
    const int mtile = blockIdx.x;
    const int lane  = threadIdx.x;
    const int arow  = lane & 15;
    const int kb    = (lane >> 4) * 2;
    const float* Abase = A + (size_t)(mtile * 16 + arow) * lda;

    v8f acc = {};
    for (int k4 = 0; k4 < K4; ++k4) {
        if ((k4 & 7) == 0 && (k4 + 8) < K4)
            __builtin_prefetch(Abase + (k4 + 8) * 4, 0, 1);   // global_prefetch_b8
        v2f a = *(const v2f*)(Abase + k4 * 4 + kb);
        v2f b = *(const v2f*)(Bf + ((size_t)(k4 * Ntiles + ntile) * 32 + lane) * 2);
        acc = __builtin_amdgcn_wmma_f32_16x16x4_f32(false, a, false, b,
                                                    (short)0, acc, false, false);
    }

    const int col = lane & 15;
    const int gn  = ntile * 16 + col;
    const int mr  = (lane >> 4) * 8;
    const float bv = bias ? bias[gn] : 0.0f;
#pragma unroll
    for (int r = 0; r < 8; ++r) {
        const int gm = mtile * 16 + mr + r;
        float v = acc[r] + bv;
        if (act) v = v / (1.0f + __expf(-v));     // silu
        if (res) v += res[(size_t)gm * ldr + gn];
        C[(size_t)gm * ldc + gn] = v;
    }
}

// ---------------------------------------------------------------------------
// Swizzle f32 weights (K x Nsrc) into WMMA B-fragment order, zero-padded.
// mode 0: direct rows; mode 1: concat remap (rows 0..247 -> 0..247,
// rows 248..495 -> 256..503, gaps zero) for Wl1 with padded [h|agg].
// ---------------------------------------------------------------------------
__global__ void prep_frag(const float* __restrict__ W, int Ksrc, int Nsrc,
                          float* __restrict__ dst, int K4, int Ntiles, int mode)
{
    const long long total = (long long)K4 * Ntiles * 64;
    long long idx = (long long)blockIdx.x * blockDim.x + threadIdx.x;
    if (idx >= total) return;
    const int i    = (int)(idx & 1);
    const int lane = (int)((idx >> 1) & 31);
    const long long fi = idx >> 6;
    const int ntile = (int)(fi % Ntiles);
    const int k4    = (int)(fi / Ntiles);
    const int k = k4 * 4 + ((lane >> 4) * 2) + i;
    const int n = ntile * 16 + (lane & 15);
    int r;
    if (mode == 0) r = k;
    else r = (k < A_DIM) ? k : ((k >= DP) ? (k - (DP - A_DIM)) : -1);
    float v = 0.0f;
    if (r >= 0 && r < Ksrc && n < Nsrc) v = W[(size_t)r * Nsrc + n];
    dst[idx] = v;
}

__global__ void pad_bias(const float* __restrict__ src, float* __restrict__ dst,
                         int n, int npad)
{
    int t = blockIdx.x * blockDim.x + threadIdx.x;
    if (t < npad) dst[t] = (t < n) ? src[t] : 0.0f;
}

__global__ void zero_strided(float* __restrict__ base, int rows, int cols, int ld)
{
    long long t = (long long)blockIdx.x * blockDim.x + threadIdx.x;
    if (t >= (long long)rows * cols) return;
    int r = (int)(t / cols), c = (int)(t % cols);
    base[(size_t)r * ld + c] = 0.0f;
}

// ---------------------------------------------------------------------------
// Lie bracket per edge: m_k = sum_z Cv[z] * h_src[Ci[z]] * h_tgt[Cj[z]].
// 1 edge per wave, 8 edges per 256-thread block.
// mode 0: atomic scatter m into agg region (cols 256.. of hcat) at row tgt.
// mode 1: bnorm = |m|^2, atomic add into bpool[tgt].
// ---------------------------------------------------------------------------
#define BE 8
__global__ __launch_bounds__(256)
void bracket_kernel(const float* __restrict__ hcat,
                    const int* __restrict__ srcI, const int* __restrict__ tgtI,
                    const float* __restrict__ Cv, const int* __restrict__ Ci,
                    const int* __restrict__ Cj, const int* __restrict__ Ck,
                    float* __restrict__ agg, float* __restrict__ bpool,
                    int E, int mode)
{
    __shared__ float sCv[NNZ_C];
    __shared__ int   sCi[NNZ_C], sCj[NNZ_C], sCk[NNZ_C];
    __shared__ float sHs[BE][DP], sHt[BE][DP], sM[BE][DP];

    const int w = threadIdx.y, lane = threadIdx.x;
    const int tid = w * 32 + lane;
    for (int z = tid; z < NNZ_C; z += 256) {
        sCv[z] = Cv[z]; sCi[z] = Ci[z]; sCj[z] = Cj[z]; sCk[z] = Ck[z];
    }

    const int e = blockIdx.x * BE + w;
    const bool active = (e < E);
    int sN = 0, tN = 0;
    if (active) { sN = srcI[e]; tN = tgtI[e]; }
#pragma unroll
    for (int i = 0; i < 8; ++i) {
        int c = lane + i * 32;
        float vs = 0.0f, vt = 0.0f;
        if (active) {
            vs = hcat[(size_t)sN * LDH + c];
            vt = hcat[(size_t)tN * LDH + c];
        }
        sHs[w][c] = vs; sHt[w][c] = vt; sM[w][c] = 0.0f;
    }
    __syncthreads();

    if (active) {
#pragma unroll 4
        for (int it = 0; it < NNZ_C / 32; ++it) {
            int z = lane + it * 32;
            float v = sCv[z] * sHs[w][sCi[z]] * sHt[w][sCj[z]];
            atomicAdd(&sM[w][sCk[z]], v);              // ds_add_f32
        }
    }
    __syncthreads();

    if (active) {
        if (mode == 0) {
#pragma unroll
            for (int i = 0; i < 8; ++i) {
                int c = lane + i * 32;                 // pad cols add 0.0 -> safe
                atomicAdd(&agg[(size_t)tN * LDH + DP + c], sM[w][c]);
            }
        } else {
            float p = 0.0f;
#pragma unroll
            for (int i = 0; i < 8; ++i) {
                float m = sM[w][lane + i * 32];
                p += m * m;
            }
            for (int off = 16; off; off >>= 1) p += __shfl_xor(p, off, 32);
            if (lane == 0) atomicAdd(&bpool[tN], p);
        }
    }
}

// killing[n] = sum_c (h@Kmat)[n,c] * h[n,c]   (one wave per node)
__global__ __launch_bounds__(256)
void killing_kernel(const float* __restrict__ t2, const float* __restrict__ hcat,
                    float* __restrict__ kil, int Nn)
{
    int node = blockIdx.x * blockDim.y + threadIdx.y;
    if (node >= Nn) return;
    int lane = threadIdx.x;
    float s = 0.0f;
#pragma unroll
    for (int i = 0; i < 8; ++i) {
        int c = lane + i * 32;
        s += t2[(size_t)node * DP + c] * hcat[(size_t)node * LDH + c];
    }
    for (int off = 16; off; off >>= 1) s += __shfl_xor(s, off, 32);
    if (lane == 0) kil[node] = s;
}

// mean-pool node feats [h(248), killing, bpool] per graph (one wave per node)
__global__ __launch_bounds__(256)
void pool_kernel(const float* __restrict__ hcat, const float* __restrict__ kil,
                 const float* __restrict__ bpool, const int* __restrict__ batch,
                 float* __restrict__ pooled, float* __restrict__ counts, int Nn)
{
    int node = blockIdx.x * blockDim.y + threadIdx.y;
    if (node >= Nn) return;
    int lane = threadIdx.x;
    int g = batch[node];
#pragma unroll
    for (int i = 0; i < 8; ++i) {
        int c = lane + i * 32;
        if (c < A_DIM) atomicAdd(&pooled[(size_t)g * DP + c], hcat[(size_t)node * LDH + c]);
    }
    if (lane == 0) {
        atomicAdd(&pooled[(size_t)g * DP + A_DIM],     kil[node]);
        atomicAdd(&pooled[(size_t)g * DP + A_DIM + 1], bpool[node]);
        atomicAdd(&counts[g], 1.0f);
    }
}

__global__ void norm_kernel(float* __restrict__ pooled, const float* __restrict__ counts)
{
    int t = blockIdx.x * blockDim.x + threadIdx.x;
    if (t < N_GRAPHS * DP) pooled[t] /= fmaxf(counts[t >> 8], 1.0f);
}

// ---------------------------------------------------------------------------
extern "C" void kernel_launch(void* const* d_in, const int* in_sizes, int n_in,
                              void* d_out, int out_size, void* d_ws, size_t ws_size,
                              hipStream_t stream)
{
    const float* x     = (const float*)d_in[0];
    const int*   eidx  = (const int*)d_in[1];
    const int*   batch = (const int*)d_in[2];
    const float* W_in1 = (const float*)d_in[3];
    const float* b_in1 = (const float*)d_in[4];
    const float* W_in2 = (const float*)d_in[5];
    const float* b_in2 = (const float*)d_in[6];
    const float* Wl1   = (const float*)d_in[7];
    const float* bl1   = (const float*)d_in[8];
    const float* Wl2   = (const float*)d_in[9];
    const float* bl2   = (const float*)d_in[10];
    const float* Cv    = (const float*)d_in[11];
    const int*   Ci    = (const int*)d_in[12];
    const int*   Cj    = (const int*)d_in[13];
    const int*   Ck    = (const int*)d_in[14];
    const float* Wo1   = (const float*)d_in[15];
    const float* bo1   = (const float*)d_in[16];
    const float* Wo2   = (const float*)d_in[17];
    const float* bo2   = (const float*)d_in[18];
    const float* Kmat  = (const float*)d_in[19];
    float* out = (float*)d_out;
    const int* srcI = eidx;
    const int* tgtI = eidx + N_EDGES;
    (void)in_sizes; (void)n_in; (void)out_size; (void)ws_size;

    char* wsb = (char*)d_ws;
    size_t off = 0;
    auto alloc = [&](size_t bytes) -> void* {
        off = (off + 255) & ~(size_t)255;
        void* p = wsb + off;
        off += bytes;
        return p;
    };
    float* hcat   = (float*)alloc((size_t)N_NODES * LDH * 4);  // [h | agg]
    float* tmp    = (float*)alloc((size_t)N_NODES * DP * 4);
    float* kil    = (float*)alloc((size_t)N_NODES * 4);
    float* bpool  = (float*)alloc((size_t)N_NODES * 4);
    float* pooled = (float*)alloc((size_t)N_GRAPHS * DP * 4);
    float* counts = (float*)alloc((size_t)N_GRAPHS * 4);
    float* og     = (float*)alloc((size_t)N_GRAPHS * DP * 4);
    float* b_in2p = (float*)alloc(DP * 4);
    float* bl2p   = (float*)alloc(2 * DP * 4);
    float* Win1f  = (float*)alloc((size_t)32  * 16 * 64 * 4);
    float* Win2f  = (float*)alloc((size_t)64  * 16 * 64 * 4);
    float* Wl1f   = (float*)alloc((size_t)2 * 128 * 16 * 64 * 4);
    float* Wl2f   = (float*)alloc((size_t)2 * 64  * 16 * 64 * 4);
    float* Kmatf  = (float*)alloc((size_t)64  * 16 * 64 * 4);
    float* Wo1f   = (float*)alloc((size_t)64  * 16 * 64 * 4);
    float* Wo2f   = (float*)alloc((size_t)64  * 1  * 64 * 4);

    auto nb = [](long long total, int tpb) { return (int)((total + tpb - 1) / tpb); };

    // --- weight fragment prep (once per launch; deterministic) ---
    prep_frag<<<nb(32LL*16*64,256),256,0,stream>>>(W_in1, IN_DIM, H_DIM, Win1f, 32, 16, 0);
    prep_frag<<<nb(64LL*16*64,256),256,0,stream>>>(W_in2, H_DIM, A_DIM, Win2f, 64, 16, 0);
    for (int l = 0; l < N_LAYERS; ++l) {
        prep_frag<<<nb(128LL*16*64,256),256,0,stream>>>(
            Wl1 + (size_t)l * 2 * A_DIM * H_DIM, 2 * A_DIM, H_DIM,
            Wl1f + (size_t)l * 128 * 16 * 64, 128, 16, 1);
        prep_frag<<<nb(64LL*16*64,256),256,0,stream>>>(
            Wl2 + (size_t)l * H_DIM * A_DIM, H_DIM, A_DIM,
            Wl2f + (size_t)l * 64 * 16 * 64, 64, 16, 0);
    }
    prep_frag<<<nb(64LL*16*64,256),256,0,stream>>>(Kmat, A_DIM, A_DIM, Kmatf, 64, 16, 0);
    prep_frag<<<nb(64LL*16*64,256),256,0,stream>>>(Wo1, A_DIM + 2, H_DIM, Wo1f, 64, 16, 0);
    prep_frag<<<nb(64LL*1*64,256),256,0,stream>>>(Wo2, H_DIM, OUT_DIM, Wo2f, 64, 1, 0);
    pad_bias<<<1,256,0,stream>>>(b_in2, b_in2p, A_DIM, DP);
    for (int l = 0; l < N_LAYERS; ++l)
        pad_bias<<<1,256,0,stream>>>(bl2 + (size_t)l * A_DIM, bl2p + (size_t)l * DP, A_DIM, DP);

    const dim3 gb(32, 4);
    const int MT = N_NODES / 16;   // 625 row-tiles

    // --- input MLP: h = silu(x@W1+b1)@W2+b2, zero-padded to 256 cols ---
    wmma_gemm_f32<<<dim3(MT,4), gb, 0, stream>>>(x,   IN_DIM, Win1f, b_in1,  nullptr, 0, tmp,  DP,  32, 16, 1);
    wmma_gemm_f32<<<dim3(MT,4), gb, 0, stream>>>(tmp, DP,     Win2f, b_in2p, nullptr, 0, hcat, LDH, 64, 16, 0);

    // --- message-passing layers ---
    for (int l = 0; l < N_LAYERS; ++l) {
        zero_strided<<<nb((long long)N_NODES*DP,256),256,0,stream>>>(hcat + DP, N_NODES, DP, LDH);
        bracket_kernel<<<N_EDGES/BE, dim3(32,BE), 0, stream>>>(
            hcat, srcI, tgtI, Cv, Ci, Cj, Ck, hcat, nullptr, N_EDGES, 0);
        wmma_gemm_f32<<<dim3(MT,4), gb, 0, stream>>>(
            hcat, LDH, Wl1f + (size_t)l * 128 * 16 * 64, bl1 + (size_t)l * H_DIM,
            nullptr, 0, tmp, DP, 128, 16, 1);
        wmma_gemm_f32<<<dim3(MT,4), gb, 0, stream>>>(
            tmp, DP, Wl2f + (size_t)l * 64 * 16 * 64, bl2p + (size_t)l * DP,
            hcat, LDH, hcat, LDH, 64, 16, 0);       // residual h += upd
    }

    // --- invariants: bracket-norm pool + Killing scalar ---
    zero_strided<<<nb((long long)N_NODES,256),256,0,stream>>>(bpool, 1, N_NODES, N_NODES);
    bracket_kernel<<<N_EDGES/BE, dim3(32,BE), 0, stream>>>(
        hcat, srcI, tgtI, Cv, Ci, Cj, Ck, nullptr, bpool, N_EDGES, 1);
    wmma_gemm_f32<<<dim3(MT,4), gb, 0, stream>>>(hcat, LDH, Kmatf, nullptr, nullptr, 0, tmp, DP, 64, 16, 0);
    killing_kernel<<<nb(N_NODES,8), dim3(32,8), 0, stream>>>(tmp, hcat, kil, N_NODES);

    // --- graph mean pooling ---
    zero_strided<<<nb((long long)N_GRAPHS*DP,256),256,0,stream>>>(pooled, 1, N_GRAPHS*DP, N_GRAPHS*DP);
    zero_strided<<<1,256,0,stream>>>(counts, 1, N_GRAPHS, N_GRAPHS);
    pool_kernel<<<nb(N_NODES,8), dim3(32,8), 0, stream>>>(hcat, kil, bpool, batch, pooled, counts, N_NODES);
    norm_kernel<<<nb((long long)N_GRAPHS*DP,256),256,0,stream>>>(pooled, counts);

    // --- output MLP (M=16 rows: one 16-row WMMA tile) ---
    wmma_gemm_f32<<<dim3(1,4), gb, 0, stream>>>(pooled, DP, Wo1f, bo1, nullptr, 0, og,  DP,      64, 16, 1);
    wmma_gemm_f32<<<dim3(1,1), gb, 0, stream>>>(og,     DP, Wo2f, bo2, nullptr, 0, out, OUT_DIM, 64, 1,  0);
}